// Loss_68693706932246
// MI455X (gfx1250) — compile-verified
//
#include <hip/hip_runtime.h>
#include <math.h>

typedef float v2f __attribute__((ext_vector_type(2)));
typedef float v8f __attribute__((ext_vector_type(8)));

#define NCLS 15
#define EPSF 1e-6f

// ---------- math helpers ----------
__device__ __forceinline__ float bce_base(float x) {
    // max(x,0) + log1p(exp(-|x|))
    float e = __expf(-fabsf(x));
    return fmaxf(x, 0.0f) + __logf(1.0f + e);
}
__device__ __forceinline__ float bce_logits(float x, float t) {
    return bce_base(x) - x * t;
}
__device__ __forceinline__ float sigmoidf_(float x) {
    return 1.0f / (1.0f + __expf(-x));
}
__device__ __forceinline__ float focal_(float x, float t) {
    float d = t - sigmoidf_(x);
    return bce_logits(x, t) * d * d;   // gamma=2, alpha=1
}
__device__ __forceinline__ float smooth_l1_(float x, float t) {
    float n = fabsf(x - t);
    return (n < 1.0f) ? 0.5f * n * n : n - 0.5f;
}

// 32-lane sum using V_WMMA_F32_16X16X4_F32.
// A[m][0] = v[m] (lanes 0-15, vgpr0), A[m][2] = v[16+m] (lanes 16-31, vgpr0),
// A[m][1]=A[m][3]=0 (vgpr1). B = all ones. D[m][n] = v[m] + v[m+16].
// C layout: lane n holds rows 0..7 (vgprs 0..7), lane 16+n holds rows 8..15.
// Sum of 8 vgprs gives half-sums; shfl_xor(16) completes the wave sum.
// Requires EXEC all-1s (callers invoke it unconditionally).
__device__ __forceinline__ float wave_sum32(float v) {
#if __has_builtin(__builtin_amdgcn_wmma_f32_16x16x4_f32)
    v2f a; a[0] = v;    a[1] = 0.0f;
    v2f b; b[0] = 1.0f; b[1] = 1.0f;
    v8f c = {0.f, 0.f, 0.f, 0.f, 0.f, 0.f, 0.f, 0.f};
    c = __builtin_amdgcn_wmma_f32_16x16x4_f32(false, a, false, b,
                                              (short)0, c, false, false);
    float s = ((c[0] + c[1]) + (c[2] + c[3])) + ((c[4] + c[5]) + (c[6] + c[7]));
    s += __shfl_xor(s, 16, 32);
    return s;
#else
    #pragma unroll
    for (int off = 16; off > 0; off >>= 1) v += __shfl_xor(v, off, 32);
    return v;
#endif
}

// giou + obb + area + smooth-l1 scores for one prediction branch
__device__ __forceinline__ void branch_scores(
    const float* __restrict__ pd,   // 13 floats: xywh[0:4] s[4:8] r[8] l[9:13]
    const float lx[4], const float la[4], float lr, const float l12[4],
    float bls, float inv_stride,
    float& siou, float& sobb, float& sarea, float& off, float& sl1)
{
    float px = pd[0], py = pd[1], pw = pd[2], ph = pd[3];
    float ax0 = px - pw * 0.5f, ay0 = py - ph * 0.5f;
    float ax1 = px + pw * 0.5f, ay1 = py + ph * 0.5f;
    float b1x0 = fminf(ax0, ax1), b1y0 = fminf(ay0, ay1);
    float b1x1 = fmaxf(ax0, ax1), b1y1 = fmaxf(ay0, ay1);
    float bx0 = lx[0] - lx[2] * 0.5f, by0 = lx[1] - lx[3] * 0.5f;
    float bx1 = lx[0] + lx[2] * 0.5f, by1 = lx[1] + lx[3] * 0.5f;
    float b2x0 = fminf(bx0, bx1), b2y0 = fminf(by0, by1);
    float b2x1 = fmaxf(bx0, bx1), b2y1 = fmaxf(by0, by1);
    float a1 = (b1x1 - b1x0) * (b1y1 - b1y0);
    float a2 = (b2x1 - b2x0) * (b2y1 - b2y0);
    float iw = fmaxf(fminf(b1x1, b2x1) - fmaxf(b1x0, b2x0), 0.f);
    float ih = fmaxf(fminf(b1y1, b2y1) - fmaxf(b1y0, b2y0), 0.f);
    float inter = iw * ih;
    float uni = a1 + a2 - inter;
    float iou = inter / (uni + EPSF);
    float ew = fmaxf(fmaxf(b1x1, b2x1) - fminf(b1x0, b2x0), 0.f);
    float eh = fmaxf(fmaxf(b1y1, b2y1) - fminf(b1y0, b2y0), 0.f);
    float enc = ew * eh;
    float giou = iou - (enc - uni) / (enc + EPSF);

    siou = bls * (1.0f - fminf(fmaxf(giou, 0.0f), 1.0f));
    sobb = 0.0f;
    #pragma unroll
    for (int i = 0; i < 4; ++i) { float d = pd[4 + i] - la[i]; sobb += d * d; }
    { float d = pd[8] - lr; sarea = d * d; }
    off = __expf(-(siou + sobb + sarea));
    sl1 = 0.0f;
    #pragma unroll
    for (int i = 0; i < 4; ++i)
        sl1 += smooth_l1_(pd[9 + i] * inv_stride, l12[i] * inv_stride);
}

// ---------- kernels ----------
__global__ void zero_acc_kernel(float* __restrict__ acc) {
    if (threadIdx.x < 9) acc[threadIdx.x] = 0.0f;
}

// one block per batch; N[b] = max(sum(flag==1) + 1e-16, 1)
__global__ __launch_bounds__(256) void count_obj_kernel(
    const float* __restrict__ label, float* __restrict__ nOut, int G)
{
    int b = blockIdx.x;
    int GG = G * G;
    const float* L = label + (size_t)b * GG * 31;
    float s = 0.0f;
    for (int c = threadIdx.x; c < GG; c += blockDim.x)
        s += (L[(size_t)c * 31 + 13] == 1.0f) ? 1.0f : 0.0f;
    __shared__ float sm[256];
    sm[threadIdx.x] = s;
    __syncthreads();
    for (int off = 128; off > 0; off >>= 1) {
        if (threadIdx.x < off) sm[threadIdx.x] += sm[threadIdx.x + off];
        __syncthreads();
    }
    if (threadIdx.x == 0) nOut[b] = fmaxf(sm[0] + 1e-16f, 1.0f);
}

__global__ __launch_bounds__(256) void loss_level_kernel(
    const float* __restrict__ p1,  const float* __restrict__ p1d,
    const float* __restrict__ p2,  const float* __restrict__ p2d,
    const float* __restrict__ label, const float* __restrict__ nPer,
    float* __restrict__ acc, int G, float stride)
{
    const int GG = G * G;
    const int total = 16 * GG;
    const float inv_stride = 1.0f / stride;
    const float img = stride * (float)G;
    const float inv_img2 = 1.0f / (img * img);

    float t[9];
    #pragma unroll
    for (int k = 0; k < 9; ++k) t[k] = 0.0f;

    // total is a multiple of 256 -> whole blocks pass/fail the bound together,
    // so waves stay convergent and WMMA below sees EXEC == all ones.
    for (int idx = blockIdx.x * 256 + threadIdx.x; idx < total;
         idx += gridDim.x * 256) {
        int b = idx / GG;
        float invN = 1.0f / nPer[b];

        const float* L = label + (size_t)idx * 31;
        float lx[4], l12[4], la[4];
        #pragma unroll
        for (int i = 0; i < 4; ++i) { lx[i] = L[i]; l12[i] = L[4 + i]; la[i] = L[8 + i]; }
        float lr = L[12], flag = L[13], mix = L[14], areaF = L[15];
        float obj   = (flag == 1.0f) ? 1.0f : 0.0f;
        float noobj = (flag == 0.0f) ? 1.0f : 0.0f;
        float fuzzy = 1.0f - obj - noobj;

        float cls[NCLS];
        float gh = L[16];
        cls[0] = L[16];
        #pragma unroll
        for (int c = 1; c < NCLS; ++c) { cls[c] = L[16 + c]; gh = fmaxf(gh, cls[c]); }

        float conf_t = obj * 0.99f + 0.005f;                  // label_conf_smooth
        float area_w = areaF + ((areaF == 0.0f) ? 1.0f : 0.0f);
        float bls = 2.0f - 1.0f * lx[2] * lx[3] * inv_img2;   // bbox_loss_scale
        float maw = mix * area_w;

        // ---- branch 1 ----
        float siou1, sobb1, sarea1, off1, sl11;
        branch_scores(p1d + (size_t)idx * 13, lx, la, lr, l12, bls, inv_stride,
                      siou1, sobb1, sarea1, off1, sl11);
        float g1 = 0.5f * (gh + off1);
        float bg_m1 = noobj + fuzzy * ((g1 < 0.3f) ? 1.0f : 0.0f) * (1.0f - fuzzy * g1);
        float fg_m1 = obj * ((g1 >= 0.3f) ? 1.0f : 0.0f);
        float foc1 = focal_(p1[(size_t)idx * 10 + 9], conf_t);
        float fg1  = fg_m1 * foc1 * mix * g1;
        float bg1  = bg_m1 * foc1 * mix;
        float iou1 = fg_m1 * siou1  * maw;
        float s1   = fg_m1 * sobb1  * maw;
        float r1   = fg_m1 * sarea1 * maw;
        float l1   = fg_m1 * bls * sl11 * maw;

        // ---- branch 2 ----
        float siou2, sobb2, sarea2, off2, sl12;
        branch_scores(p2d + (size_t)idx * 13, lx, la, lr, l12, bls, inv_stride,
                      siou2, sobb2, sarea2, off2, sl12);
        float g2 = 0.5f * (gh + off2);
        float bg_m2 = noobj + fuzzy * ((g2 < 0.3f) ? 1.0f : 0.0f) * (1.0f - fuzzy * g2);
        float fg_m2 = obj * ((g2 >= 0.3f) ? 1.0f : 0.0f);
        const float* P2 = p2 + (size_t)idx * 25;
        float foc2 = focal_(P2[9], conf_t);
        float fg2 = fg_m2 * foc2 * mix * g2;
        float bg2 = bg_m2 * foc2 * mix;

        float pos2 = 0.0f, neg2 = 0.0f, cls2 = 0.0f, wsum = 0.0f;
        #pragma unroll
        for (int c = 0; c < NCLS; ++c) {
            float x  = P2[10 + c];
            float nz = (cls[c] != 0.0f) ? 1.0f : 0.0f;
            float tsm = nz * 0.99f + (0.01f / 15.0f);         // label_cls_smooth
            float e = __expf(-fabsf(x));
            float base = fmaxf(x, 0.0f) + __logf(1.0f + e);   // bce minus x*t term
            float sg = (x >= 0.0f) ? 1.0f / (1.0f + e) : e / (1.0f + e);
            pos2 += nz * (base - x * (tsm * off2));
            neg2 += (1.0f - nz) * (base - x * tsm);
            cls2 += (base - x * tsm);
            wsum += nz * sg;
        }
        pos2 *= obj * maw;
        neg2 *= obj * maw;
        cls2 *= fg_m2 * maw;

        float wg = 0.5f * (wsum + gh);
        float mawg = maw * wg;
        float iou2 = fg_m2 * siou2  * mawg;
        float s2   = fg_m2 * sobb2  * mawg;
        float r2   = fg_m2 * sarea2 * mawg;
        float l2   = fg_m2 * bls * sl12 * mawg;

        t[0] += (fg1 + fg2) * invN;
        t[1] += (bg1 + bg2) * invN;
        t[2] += pos2 * invN;
        t[3] += neg2 * invN;
        t[4] += (iou1 + iou2) * invN;
        t[5] += cls2 * invN;
        t[6] += (s1 + s2) * invN;
        t[7] += (r1 + r2) * invN;
        t[8] += (l1 + l2) * invN;
    }

    // wave reduction via v_wmma_f32_16x16x4_f32 (EXEC all ones here)
    __shared__ float red[8][9];
    int wave = threadIdx.x >> 5;
    int lane = threadIdx.x & 31;
    #pragma unroll
    for (int k = 0; k < 9; ++k) {
        float s = wave_sum32(t[k]);
        if (lane == 0) red[wave][k] = s;
    }
    __syncthreads();
    if (threadIdx.x < 9) {
        float s = 0.0f;
        #pragma unroll
        for (int w = 0; w < 8; ++w) s += red[w][threadIdx.x];
        atomicAdd(&acc[threadIdx.x], s);
    }
}

__global__ void finalize_kernel(const float* __restrict__ acc,
                                float* __restrict__ out)
{
    if (threadIdx.x == 0 && blockIdx.x == 0) {
        const float invB = 1.0f / 16.0f;
        float fg  = acc[0] * invB;
        float bg  = acc[1] * invB;
        float pos = 28.0f * acc[2] * invB;   // 2 * (NUM_CLASS-1)
        float neg = 2.0f  * acc[3] * invB;
        float iou = 0.5f  * acc[4] * invB;
        float cls = acc[5] * invB;
        float s   = 0.5f  * acc[6] * invB;
        float r   = 8.0f  * acc[7] * invB;   // 16 / 2
        float l   = 0.1f  * acc[8] * invB;   // 0.2 / 2
        out[0] = fg + bg + iou + s + r + pos + neg + l;  // cls excluded
        out[1] = fg;  out[2] = bg;  out[3] = pos; out[4] = neg;
        out[5] = iou; out[6] = cls; out[7] = s;   out[8] = r; out[9] = l;
    }
}

extern "C" void kernel_launch(void* const* d_in, const int* in_sizes, int n_in,
                              void* d_out, int out_size, void* d_ws, size_t ws_size,
                              hipStream_t stream)
{
    (void)in_sizes; (void)n_in; (void)out_size; (void)ws_size;
    // d_in dict order: [p1_s,p1d_s,p2_s,p2d_s,label_s, p1_m,...,label_m,
    //                   p1_l,...,label_l, epoch, i]
    float* acc = (float*)d_ws;      // 9 accumulators
    float* N   = acc + 16;          // 3 levels x 16 batches

    zero_acc_kernel<<<1, 32, 0, stream>>>(acc);

    count_obj_kernel<<<16, 256, 0, stream>>>((const float*)d_in[4],  N + 0,  128);
    count_obj_kernel<<<16, 256, 0, stream>>>((const float*)d_in[9],  N + 16, 64);
    count_obj_kernel<<<16, 256, 0, stream>>>((const float*)d_in[14], N + 32, 32);

    auto launch_level = [&](int base, int G, float stride, float* nPer) {
        int total = 16 * G * G;
        int blocks = (total + 255) / 256;
        if (blocks > 1024) blocks = 1024;
        loss_level_kernel<<<blocks, 256, 0, stream>>>(
            (const float*)d_in[base + 0], (const float*)d_in[base + 1],
            (const float*)d_in[base + 2], (const float*)d_in[base + 3],
            (const float*)d_in[base + 4], nPer, acc, G, stride);
    };
    launch_level(0,  128, 8.0f,  N + 0);
    launch_level(5,  64,  16.0f, N + 16);
    launch_level(10, 32,  32.0f, N + 32);

    finalize_kernel<<<1, 32, 0, stream>>>(acc, (float*)d_out);
}